// GAT_79577154060349
// MI455X (gfx1250) — compile-verified
//
#include <hip/hip_runtime.h>
#include <math.h>

// ---------------------------------------------------------------------------
// GAT (2-layer) for MI455X / gfx1250, wave32.
//  - Dense GEMMs use V_WMMA_F32_16X16X4_F32 (fp32 WMMA -> matches reference
//    precision), W panel staged in LDS (W1=64KB, W2=32KB; WGP has 320KB).
//  - Edge pipeline (logits / segment softmax / scatter) is memory bound:
//    ~4-5 GB total traffic -> ~0.2ms at 23.3 TB/s; implemented with fp32
//    atomics (L2 atomic units) and coalesced per-wave feature striding.
// ---------------------------------------------------------------------------

typedef __attribute__((ext_vector_type(2))) float v2f;
typedef __attribute__((ext_vector_type(8))) float v8f;

// ---- order-preserving float<->uint key for atomicMax-based segment max ----
__device__ __forceinline__ unsigned fkey(float f) {
  unsigned u = __float_as_uint(f);
  return (u & 0x80000000u) ? ~u : (u | 0x80000000u);
}
__device__ __forceinline__ float funkey(unsigned k) {
  unsigned u = (k & 0x80000000u) ? (k & 0x7fffffffu) : ~k;
  return __uint_as_float(u);
}

// ---------------------------------------------------------------------------
// GEMM: Out[M x NCOLS] = A[M x 128] * W[128 x NCOLS], fp32 WMMA 16x16x4.
// One wave per 16x16 output tile; block covers one 16-row strip and all NCOLS.
// ---------------------------------------------------------------------------
template <int NCOLS>
__global__ __launch_bounds__(32 * (NCOLS / 16))
void gemm_wmma_f32(const float* __restrict__ A, const float* __restrict__ Wg,
                   float* __restrict__ Out, int M) {
  constexpr int K = 128;
  constexpr int NT = NCOLS / 16;           // waves per block = n-tiles
  __shared__ float Ws[K * NCOLS];          // 64KB (NCOLS=128) / 32KB (64)

  const int tid = threadIdx.x;
  for (int i = tid; i < K * NCOLS; i += 32 * NT) Ws[i] = Wg[i];
  __syncthreads();

  const int wave  = tid >> 5;              // n-tile index
  const int lane  = tid & 31;
  const int row16 = blockIdx.x << 4;
  const int r     = lane & 15;
  const int kb    = (lane >> 4) << 1;      // 0 (lanes 0-15) or 2 (lanes 16-31)
  const int col   = (wave << 4) + r;

  int arow = row16 + r;
  if (arow >= M) arow = M - 1;             // clamp: keeps EXEC all-ones for WMMA
  const float* __restrict__ ap = A + (size_t)arow * K;

  v8f acc = {};
#pragma unroll
  for (int k = 0; k < K; k += 4) {
    v2f a, b;
    a.x = ap[k + kb];
    a.y = ap[k + kb + 1];
    b.x = Ws[(k + kb) * NCOLS + col];
    b.y = Ws[(k + kb + 1) * NCOLS + col];
    acc = __builtin_amdgcn_wmma_f32_16x16x4_f32(false, a, false, b,
                                                (short)0, acc, false, false);
  }

  const int orow0 = row16 + ((lane >> 4) << 3);  // lanes 16-31 -> rows +8
  const int ocol  = (wave << 4) + r;
#pragma unroll
  for (int i = 0; i < 8; ++i) {
    int orow = orow0 + i;
    if (orow < M) Out[(size_t)orow * NCOLS + ocol] = acc[i];
  }
}

// ---------------------------------------------------------------------------
// Self-loop attr = mean of incoming edge weights (PyG fill_value='mean').
// ---------------------------------------------------------------------------
__global__ void k_selfloop_stats(const int* __restrict__ dst,
                                 const float* __restrict__ ew,
                                 float* __restrict__ deg,
                                 float* __restrict__ loopw, int E) {
  int e = blockIdx.x * blockDim.x + threadIdx.x;
  if (e >= E) return;
  int d = dst[e];
  atomicAdd(&deg[d], 1.0f);
  atomicAdd(&loopw[d], ew[e]);
}

__global__ void k_selfloop_div(float* __restrict__ loopw,
                               const float* __restrict__ deg, int N) {
  int n = blockIdx.x * blockDim.x + threadIdx.x;
  if (n >= N) return;
  loopw[n] = loopw[n] / fmaxf(deg[n], 1.0f);
}

// ---------------------------------------------------------------------------
// Per-node attention coefficients: s[n,h] = <h[n,h,:], a_s[h,:]> (and a_d).
// ---------------------------------------------------------------------------
template <int H>
__global__ void k_node_coef(const float* __restrict__ hfeat,
                            const float* __restrict__ a_s,
                            const float* __restrict__ a_d,
                            float* __restrict__ scoef,
                            float* __restrict__ dcoef, int N) {
  int n = blockIdx.x * blockDim.x + threadIdx.x;
  if (n >= N) return;
  const float* hr = hfeat + (size_t)n * (H * 64);
#pragma unroll
  for (int h = 0; h < H; ++h) {
    float ss = 0.f, dd = 0.f;
#pragma unroll 8
    for (int c = 0; c < 64; ++c) {
      float v = hr[h * 64 + c];
      ss += v * a_s[h * 64 + c];
      dd += v * a_d[h * 64 + c];
    }
    scoef[n * H + h] = ss;
    dcoef[n * H + h] = dd;
  }
}

// cst[h] = sum_c We[h*64+c] * ae[h*64+c]  (edge attr is scalar per edge)
template <int H>
__global__ void k_edge_const(const float* __restrict__ We,
                             const float* __restrict__ ae,
                             float* __restrict__ cst) {
  int h = threadIdx.x;
  if (h < H) {
    float s = 0.f;
    for (int c = 0; c < 64; ++c) s += We[h * 64 + c] * ae[h * 64 + c];
    cst[h] = s;
  }
}

// ---------------------------------------------------------------------------
// Edge logits + leaky-relu + segment max (atomicMax on ordered uint key).
// Edges e < E come from edge_index; e >= E are self-loops.
// ---------------------------------------------------------------------------
template <int H>
__global__ void k_edge_logit_max(const int* __restrict__ src,
                                 const int* __restrict__ dst,
                                 const float* __restrict__ ew,
                                 const float* __restrict__ loopw,
                                 const float* __restrict__ scoef,
                                 const float* __restrict__ dcoef,
                                 const float* __restrict__ cst,
                                 float* __restrict__ albuf,
                                 unsigned* __restrict__ mkey, int E, int ET) {
  int e = blockIdx.x * blockDim.x + threadIdx.x;
  if (e >= ET) return;
  int s, d;
  float w;
  if (e < E) { s = src[e]; d = dst[e]; w = ew[e]; }
  else       { s = d = e - E;          w = loopw[s]; }
#pragma unroll
  for (int h = 0; h < H; ++h) {
    float al = scoef[s * H + h] + dcoef[d * H + h] + w * cst[h];
    al = (al >= 0.f) ? al : 0.2f * al;                 // leaky_relu(0.2)
    albuf[(size_t)e * H + h] = al;
    atomicMax(&mkey[d * H + h], fkey(al));
  }
}

// exp(al - max) and segment sum
template <int H>
__global__ void k_edge_expsum(const int* __restrict__ dst,
                              const float* __restrict__ albuf,
                              const unsigned* __restrict__ mkey,
                              float* __restrict__ zbuf,
                              float* __restrict__ ssum, int E, int ET) {
  int e = blockIdx.x * blockDim.x + threadIdx.x;
  if (e >= ET) return;
  int d = (e < E) ? dst[e] : (e - E);
#pragma unroll
  for (int h = 0; h < H; ++h) {
    float z = expf(albuf[(size_t)e * H + h] - funkey(mkey[d * H + h]));
    zbuf[(size_t)e * H + h] = z;
    atomicAdd(&ssum[d * H + h], z);
  }
}

// Weighted scatter: agg[dst, f] += h[src, f] * (z / ssum)[head(f)].
// One wave32 per edge; lanes stride the F features (coalesced).
template <int F>  // F = H*64
__global__ void k_edge_aggregate(const int* __restrict__ src,
                                 const int* __restrict__ dst,
                                 const float* __restrict__ hfeat,
                                 const float* __restrict__ zbuf,
                                 const float* __restrict__ ssum,
                                 float* __restrict__ agg, int E, int ET) {
  constexpr int H = F / 64;
  int gt = blockIdx.x * blockDim.x + threadIdx.x;
  int e = gt >> 5;
  int lane = gt & 31;
  if (e >= ET) return;
  int s, d;
  if (e < E) { s = src[e]; d = dst[e]; }
  else       { s = d = e - E; }
  float wc[H];
#pragma unroll
  for (int h = 0; h < H; ++h)
    wc[h] = zbuf[(size_t)e * H + h] / ssum[d * H + h];
  const float* hs = hfeat + (size_t)s * F;
  float* ad = agg + (size_t)d * F;
#pragma unroll
  for (int f = lane; f < F; f += 32)
    atomicAdd(&ad[f], hs[f] * wc[f >> 6]);
}

// out[i] = act(out[i] + bias[i % F]); ELU for layer-1 output, plain for final.
template <int F, bool ELU>
__global__ void k_bias_act(float* __restrict__ a, const float* __restrict__ b,
                           int n) {
  int i = blockIdx.x * blockDim.x + threadIdx.x;
  if (i >= n) return;
  float v = a[i] + b[i & (F - 1)];
  if (ELU) v = (v > 0.f) ? v : expm1f(v);
  a[i] = v;
}

// ---------------------------------------------------------------------------
extern "C" void kernel_launch(void* const* d_in, const int* in_sizes, int n_in,
                              void* d_out, int out_size, void* d_ws,
                              size_t ws_size, hipStream_t stream) {
  const float* x   = (const float*)d_in[0];
  const int*  eidx = (const int*)d_in[1];
  const float* ew  = (const float*)d_in[2];
  const float* W1  = (const float*)d_in[3];
  const float* We1 = (const float*)d_in[4];
  const float* as1 = (const float*)d_in[5];
  const float* ad1 = (const float*)d_in[6];
  const float* ae1 = (const float*)d_in[7];
  const float* b1  = (const float*)d_in[8];
  const float* W2  = (const float*)d_in[9];
  const float* We2 = (const float*)d_in[10];
  const float* as2 = (const float*)d_in[11];
  const float* ad2 = (const float*)d_in[12];
  const float* ae2 = (const float*)d_in[13];
  const float* b2  = (const float*)d_in[14];
  float* out = (float*)d_out;

  const int N  = in_sizes[0] / 128;  // 100000
  const int E  = in_sizes[2];        // 1600000
  const int ET = E + N;
  const int* srcv = eidx;            // edge_index[0]
  const int* dstv = eidx + E;        // edge_index[1]

  // ---- workspace layout (~134 MB of floats) ----
  float* p = (float*)d_ws;
  float* loopw = p; p += N;                    // self-loop attrs (and wsum tmp)
  float* deg   = p; p += N;
  float* scoef = p; p += 2 * (size_t)N;
  float* dcoef = p; p += 2 * (size_t)N;
  float* cst   = p; p += 8;
  float* albuf = p; p += 2 * (size_t)ET;
  float* zbuf  = p; p += 2 * (size_t)ET;
  unsigned* mkey = (unsigned*)p; p += 2 * (size_t)N;
  float* ssum  = p; p += 2 * (size_t)N;
  float* hfeat = p; p += (size_t)N * 128;      // h1, then h2
  float* agg   = p; p += (size_t)N * 128;      // agg1, then elu input to layer2

  const int T = 256;
  auto cdiv = [](long long a, long long b) { return (int)((a + b - 1) / b); };

  // ---- self loops ----
  hipMemsetAsync(deg, 0, (size_t)N * 4, stream);
  hipMemsetAsync(loopw, 0, (size_t)N * 4, stream);
  k_selfloop_stats<<<cdiv(E, T), T, 0, stream>>>(dstv, ew, deg, loopw, E);
  k_selfloop_div<<<cdiv(N, T), T, 0, stream>>>(loopw, deg, N);

  // ---- layer 1 (H=2, C=64) ----
  gemm_wmma_f32<128><<<cdiv(N, 16), 256, 0, stream>>>(x, W1, hfeat, N);
  k_node_coef<2><<<cdiv(N, T), T, 0, stream>>>(hfeat, as1, ad1, scoef, dcoef, N);
  k_edge_const<2><<<1, 32, 0, stream>>>(We1, ae1, cst);
  hipMemsetAsync(mkey, 0, 2 * (size_t)N * 4, stream);   // key 0 == -inf
  hipMemsetAsync(ssum, 0, 2 * (size_t)N * 4, stream);
  hipMemsetAsync(agg, 0, (size_t)N * 128 * 4, stream);
  k_edge_logit_max<2><<<cdiv(ET, T), T, 0, stream>>>(srcv, dstv, ew, loopw,
                                                     scoef, dcoef, cst, albuf,
                                                     mkey, E, ET);
  k_edge_expsum<2><<<cdiv(ET, T), T, 0, stream>>>(dstv, albuf, mkey, zbuf,
                                                  ssum, E, ET);
  k_edge_aggregate<128><<<cdiv((long long)ET * 32, T), T, 0, stream>>>(
      srcv, dstv, hfeat, zbuf, ssum, agg, E, ET);
  k_bias_act<128, true><<<cdiv((long long)N * 128, T), T, 0, stream>>>(
      agg, b1, N * 128);  // agg now holds elu(h1_out + b1) = layer-2 input

  // ---- layer 2 (H=1, C=64) ----
  gemm_wmma_f32<64><<<cdiv(N, 16), 128, 0, stream>>>(agg, W2, hfeat, N);
  k_node_coef<1><<<cdiv(N, T), T, 0, stream>>>(hfeat, as2, ad2, scoef, dcoef, N);
  k_edge_const<1><<<1, 32, 0, stream>>>(We2, ae2, cst);
  hipMemsetAsync(mkey, 0, (size_t)N * 4, stream);
  hipMemsetAsync(ssum, 0, (size_t)N * 4, stream);
  hipMemsetAsync(out, 0, (size_t)N * 64 * 4, stream);
  k_edge_logit_max<1><<<cdiv(ET, T), T, 0, stream>>>(srcv, dstv, ew, loopw,
                                                     scoef, dcoef, cst, albuf,
                                                     mkey, E, ET);
  k_edge_expsum<1><<<cdiv(ET, T), T, 0, stream>>>(dstv, albuf, mkey, zbuf,
                                                  ssum, E, ET);
  k_edge_aggregate<64><<<cdiv((long long)ET * 32, T), T, 0, stream>>>(
      srcv, dstv, hfeat, zbuf, ssum, out, E, ET);
  k_bias_act<64, false><<<cdiv((long long)N * 64, T), T, 0, stream>>>(
      out, b2, N * 64);
}